// PManifold_601295421536
// MI455X (gfx1250) — compile-verified
//
#include <hip/hip_runtime.h>
#include <math.h>

#define KK 64
#define DD 10
#define NN 1024
#define EPS_ 1e-5f

typedef __attribute__((ext_vector_type(2))) float v2f;
typedef __attribute__((ext_vector_type(8))) float v8f;

__launch_bounds__(256)
__global__ void pmanifold_kernel(const float* __restrict__ dgm_all,
                                 const float* __restrict__ theta_all,
                                 float* __restrict__ out) {
  const int bc = blockIdx.x;          // [0, 64): (b, c)
  const int b  = bc >> 1;
  const int c  = bc & 1;
  const int t  = threadIdx.x;         // [0, 256)

  __shared__ float p0s[NN], p1s[NN], pps[NN];   // parametrized points
  __shared__ float thL[KK * 16];                // theta, padded to 16 cols (WMMA B)
  __shared__ float thn2[KK];                    // sum_d theta[row][d]^2
  __shared__ float ssL[KK * KK];                // SS[row][k2]
  __shared__ float a0p[4][KK], a1p[4][KK];      // partial sums of s*p0, s*p1
  __shared__ float sbuf[KK * 16];               // matmul result S[k2][d]

  // ---- Phase 0: stage points and theta ----
  const float* dgm = dgm_all + ((size_t)(b * 2 + c)) * NN * 2;
  for (int i = t; i < NN; i += 256) {
    float d0 = dgm[i * 2 + 0], d1 = dgm[i * 2 + 1];
    float inv = 1.0f / (1.0f + sqrtf(1.0f + d0 * d0 + d1 * d1));
    float p0 = d0 * inv, p1 = d1 * inv;
    p0s[i] = p0; p1s[i] = p1; pps[i] = p0 * p0 + p1 * p1;
  }
  if (t < KK) {
    const float* th = theta_all + ((size_t)(c * KK + t)) * DD;
    float s2 = 0.f;
    for (int d = 0; d < DD; ++d) { float v = th[d]; thL[t * 16 + d] = v; s2 += v * v; }
    for (int d = DD; d < 16; ++d) thL[t * 16 + d] = 0.f;
    thn2[t] = s2;
  }
  __syncthreads();

  // ---- Phase 1: scalar arctanh scales; build SS[row][k2] and corrections ----
  const int k2 = t & 63;
  const int q  = t >> 6;                // which 16-row band this thread owns
  float pp[16], P0[16], P1[16];
  for (int j = 0; j < 16; ++j) {        // n' = (n2 & 15)*64 + k2, independent of row
    int np = j * 64 + k2;
    pp[j] = pps[np]; P0[j] = p0s[np]; P1[j] = p1s[np];
  }
  float a0 = 0.f, a1 = 0.f;
  for (int rr = 0; rr < 16; ++rr) {
    int row = q * 16 + rr;              // row = n2 >> 4
    float th0 = thL[row * 16 + 0], th1 = thL[row * 16 + 1], tn2 = thn2[row];
    float ssacc = 0.f;
    for (int j = 0; j < 16; ++j) {
      float nx2 = pp[j] + 2.0f * (P0[j] * th0 + P1[j] * th1) + tn2; // ||x||^2
      float n   = sqrtf(nx2 + 1e-12f);
      float nc  = fminf(fmaxf(n, 1e-12f), 1.0f - EPS_);
      float s   = 0.5f * __logf((1.0f + nc) / (1.0f - nc)) / n;     // arctanh(nc)/n
      ssacc += s;
      a0 += s * P0[j];
      a1 += s * P1[j];
    }
    ssL[row * 64 + k2] = ssacc;         // unique writer per (row,k2)
  }
  a0p[q][k2] = a0; a1p[q][k2] = a1;
  __syncthreads();

  // ---- Phase 2: S = SS(64x64) x theta(64x16) via v_wmma_f32_16x16x4_f32 ----
  const int wave = t >> 5;
  const int lane = t & 31;
  if (wave < 4) {                       // wave-uniform branch: EXEC all-1s inside
    const int mt   = wave;              // M tile (k2 block of 16)
    const int m    = lane & 15;         // A row / B,C column
    const int half = lane >> 4;
    v8f acc = {};
    for (int kc = 0; kc < 16; ++kc) {
      int r0 = kc * 4 + half * 2;       // A: VGPR0 -> K rows {0,2}, VGPR1 -> {1,3}
      v2f av, bv;
      av.x = ssL[(r0 + 0) * 64 + (mt * 16 + m)];
      av.y = ssL[(r0 + 1) * 64 + (mt * 16 + m)];
      bv.x = thL[(r0 + 0) * 16 + m];
      bv.y = thL[(r0 + 1) * 16 + m];
      acc = __builtin_amdgcn_wmma_f32_16x16x4_f32(
          /*neg_a=*/false, av, /*neg_b=*/false, bv,
          /*c_mod=*/(short)0, acc, /*reuse_a=*/false, /*reuse_b=*/false);
    }
    for (int v = 0; v < 8; ++v) {       // C/D: lane l, VGPR v -> M = 8*(l>>4)+v, N = l&15
      int mm = half * 8 + v;
      sbuf[(mt * 16 + mm) * 16 + m] = acc[v];
    }
  }
  __syncthreads();

  // ---- Phase 3: corrections + chart(exp_map0(S)) epilogue ----
  if (t < KK) {
    int k = t;
    float s[DD];
    for (int d = 0; d < DD; ++d) s[d] = sbuf[k * 16 + d];
    s[0] += a0p[0][k] + a0p[1][k] + a0p[2][k] + a0p[3][k];
    s[1] += a1p[0][k] + a1p[1][k] + a1p[2][k] + a1p[3][k];
    float ss2 = 1e-12f;
    for (int d = 0; d < DD; ++d) ss2 += s[d] * s[d];
    float n     = sqrtf(ss2);
    float tn    = tanhf(n);
    float scale = tn / n;                       // y = scale * s
    float y2    = scale * scale * (ss2 - 1e-12f);
    float sq    = fminf(fmaxf(y2, 0.f), 1.0f - EPS_);
    float cs    = 2.0f * scale / (1.0f - sq);   // chart: 2y/(1-||y||^2)
    float* o = out + ((size_t)((b * 2 + c) * KK + k)) * DD;
    for (int d = 0; d < DD; ++d) o[d] = cs * s[d];
  }
}

extern "C" void kernel_launch(void* const* d_in, const int* in_sizes, int n_in,
                              void* d_out, int out_size, void* d_ws, size_t ws_size,
                              hipStream_t stream) {
  (void)in_sizes; (void)n_in; (void)out_size; (void)d_ws; (void)ws_size;
  const float* inputs = (const float*)d_in[0];  // [32, 2, 1024, 2] f32
  const float* theta  = (const float*)d_in[1];  // [2, 64, 10] f32
  float* out = (float*)d_out;                   // [32, 2, 64, 10] f32
  pmanifold_kernel<<<dim3(64), dim3(256), 0, stream>>>(inputs, theta, out);
}